// GFLayer_14998025797944
// MI455X (gfx1250) — compile-verified
//
#include <hip/hip_runtime.h>
#include <math.h>

typedef __attribute__((ext_vector_type(2))) float v2f;
typedef __attribute__((ext_vector_type(8))) float v8f;

#define En   16
#define Bn   32
#define Nn   2048
#define HIDn 128
#define Fn   16
#define Sn   15

// =====================================================================
// Kernel 1: h1=relu(Xin@W1+b1); h2=relu(h1@W2+b2); h3=h2@W3+b3; Q/K/V.
// All three GEMMs via V_WMMA_F32_16X16X4_F32 chains (fp32, matches ref).
// grid (Nn/64, Bn, Sn), block 128 (4 waves, one 16-row tile per wave).
// =====================================================================
__global__ __launch_bounds__(128) void k1_mlp_qkv(
    const float* __restrict__ X,
    const float* __restrict__ W1, const float* __restrict__ b1,
    const float* __restrict__ W2, const float* __restrict__ b2,
    const float* __restrict__ W3, const float* __restrict__ b3,
    const float* __restrict__ Wq, const float* __restrict__ bq,
    const float* __restrict__ Wk, const float* __restrict__ bk,
    const float* __restrict__ Wv, const float* __restrict__ bv,
    float* __restrict__ Qws, float* __restrict__ Kws, float* __restrict__ Vws)
{
    __shared__ float buf[4][16 * HIDn];   // h1 then h2 (reused in place)
    __shared__ float buf3[4][16 * Fn];    // h3

    const int w    = threadIdx.x >> 5;
    const int lane = threadIdx.x & 31;
    const int half = lane >> 4;           // 0: lanes 0-15, 1: lanes 16-31
    const int lrow = lane & 15;
    const int s  = blockIdx.z;
    const int b  = blockIdx.y;
    const int n0 = blockIdx.x * 64 + w * 16;

    const float* W1g = W1 + s * (Sn * HIDn);
    const float* W2g = W2 + (size_t)s * HIDn * HIDn;
    const float* W3g = W3 + s * (HIDn * Fn);

    // ---------- stage 1: h1 = relu(Xin(16x15, zero-padded to 16) @ W1 + b1)
    v2f a1[4];
    {
        const float* xr = X + ((size_t)(b * Nn + n0 + lrow)) * En;
#pragma unroll
        for (int k = 0; k < 4; ++k) {
            int c0 = 4 * k + 2 * half;
            v2f t;
            t.x = (c0     < Sn) ? xr[c0]     : 0.0f;   // col 15 is padding
            t.y = (c0 + 1 < Sn) ? xr[c0 + 1] : 0.0f;
            a1[k] = t;
        }
    }
#pragma unroll
    for (int jt = 0; jt < 8; ++jt) {
        float bvv = b1[s * HIDn + jt * 16 + lrow];
        v8f c;
#pragma unroll
        for (int i = 0; i < 8; ++i) c[i] = bvv;
#pragma unroll
        for (int k = 0; k < 4; ++k) {
            int r0 = 4 * k + 2 * half;
            v2f bf;
            bf.x = (r0     < Sn) ? W1g[(size_t)r0 * HIDn       + jt * 16 + lrow] : 0.0f;
            bf.y = (r0 + 1 < Sn) ? W1g[(size_t)(r0 + 1) * HIDn + jt * 16 + lrow] : 0.0f;
            c = __builtin_amdgcn_wmma_f32_16x16x4_f32(false, a1[k], false, bf,
                                                      (short)0, c, false, false);
        }
#pragma unroll
        for (int r = 0; r < 8; ++r)
            buf[w][(r + 8 * half) * HIDn + jt * 16 + lrow] = fmaxf(c[r], 0.0f);
    }
    __syncthreads();

    // ---------- stage 2: h2 = relu(h1 @ W2 + b2)
    v2f a2[32];
#pragma unroll
    for (int k = 0; k < 32; ++k) {
        v2f t;
        t.x = buf[w][lrow * HIDn + 4 * k + 2 * half];
        t.y = buf[w][lrow * HIDn + 4 * k + 2 * half + 1];
        a2[k] = t;
    }
    __syncthreads();   // all A-fragments in regs; safe to overwrite buf
    for (int jt = 0; jt < 8; ++jt) {
        float bvv = b2[s * HIDn + jt * 16 + lrow];
        v8f c;
#pragma unroll
        for (int i = 0; i < 8; ++i) c[i] = bvv;
#pragma unroll
        for (int k = 0; k < 32; ++k) {
            int r0 = 4 * k + 2 * half;
            v2f bf;
            bf.x = W2g[(size_t)r0 * HIDn       + jt * 16 + lrow];
            bf.y = W2g[(size_t)(r0 + 1) * HIDn + jt * 16 + lrow];
            c = __builtin_amdgcn_wmma_f32_16x16x4_f32(false, a2[k], false, bf,
                                                      (short)0, c, false, false);
        }
#pragma unroll
        for (int r = 0; r < 8; ++r)
            buf[w][(r + 8 * half) * HIDn + jt * 16 + lrow] = fmaxf(c[r], 0.0f);
    }
    __syncthreads();

    // ---------- stage 3: h3 = h2 @ W3 + b3  (single 16x16 output tile)
    v2f a3[32];
#pragma unroll
    for (int k = 0; k < 32; ++k) {
        v2f t;
        t.x = buf[w][lrow * HIDn + 4 * k + 2 * half];
        t.y = buf[w][lrow * HIDn + 4 * k + 2 * half + 1];
        a3[k] = t;
    }
    {
        float bvv = b3[s * Fn + lrow];
        v8f c;
#pragma unroll
        for (int i = 0; i < 8; ++i) c[i] = bvv;
#pragma unroll
        for (int k = 0; k < 32; ++k) {
            int r0 = 4 * k + 2 * half;
            v2f bf;
            bf.x = W3g[r0 * Fn       + lrow];
            bf.y = W3g[(r0 + 1) * Fn + lrow];
            c = __builtin_amdgcn_wmma_f32_16x16x4_f32(false, a3[k], false, bf,
                                                      (short)0, c, false, false);
        }
#pragma unroll
        for (int r = 0; r < 8; ++r)
            buf3[w][(r + 8 * half) * Fn + lrow] = c[r];
    }
    __syncthreads();

    // ---------- Q/K/V projections (16 -> 2), one row per lane (lanes 0-15)
    if (lane < 16) {
        const int n = n0 + lrow;
        const float* h = &buf3[w][lrow * Fn];
        float q0 = bq[s * 2], q1 = bq[s * 2 + 1];
        float k0 = bk[s * 2], k1 = bk[s * 2 + 1];
        float v0 = bv[s * 2], v1 = bv[s * 2 + 1];
#pragma unroll
        for (int i = 0; i < Fn; ++i) {
            float hv = h[i];
            q0 = fmaf(hv, Wq[(s * Fn + i) * 2 + 0], q0);
            q1 = fmaf(hv, Wq[(s * Fn + i) * 2 + 1], q1);
            k0 = fmaf(hv, Wk[(s * Fn + i) * 2 + 0], k0);
            k1 = fmaf(hv, Wk[(s * Fn + i) * 2 + 1], k1);
            v0 = fmaf(hv, Wv[(s * Fn + i) * 2 + 0], v0);
            v1 = fmaf(hv, Wv[(s * Fn + i) * 2 + 1], v1);
        }
        const size_t base = ((size_t)(s * Bn + b) * Nn + n) * 2;
        Qws[base] = q0; Qws[base + 1] = q1;
        Kws[base] = k0; Kws[base + 1] = k1;
        Vws[base] = v0; Vws[base + 1] = v1;
    }
}

// =====================================================================
// Kernel 2: per-(s,b) full attention (DV=2) + epilogue + alpha reduction.
// grid Sn*Bn, block 256; K,V staged in LDS; 8 rows per thread.
// =====================================================================
__global__ __launch_bounds__(256) void k2_attn(
    const float* __restrict__ X,
    const float* __restrict__ Qws, const float* __restrict__ Kws,
    const float* __restrict__ Vws,
    const float* __restrict__ Wo, const float* __restrict__ bo,
    float* __restrict__ Zout, float* __restrict__ alpha_ws)
{
    __shared__ float2 Ks[Nn];
    __shared__ float2 Vs[Nn];
    __shared__ float  red[256];

    const int sb  = blockIdx.x;            // s*Bn + b
    const int s   = sb / Bn;
    const int b   = sb % Bn;
    const int tid = threadIdx.x;
    const size_t qkvbase = (size_t)sb * Nn * 2;

    const float2* Kg = (const float2*)(Kws + qkvbase);
    const float2* Vg = (const float2*)(Vws + qkvbase);
    for (int m = tid; m < Nn; m += 256) { Ks[m] = Kg[m]; Vs[m] = Vg[m]; }
    __syncthreads();

    const float scale = 0.70710678118654752440f;   // 1/sqrt(DV=2)
    float q0[8], q1[8], mx[8], se[8], o0[8], o1[8];
#pragma unroll
    for (int r = 0; r < 8; ++r) {
        int n = r * 256 + tid;
        q0[r] = Qws[qkvbase + (size_t)n * 2];
        q1[r] = Qws[qkvbase + (size_t)n * 2 + 1];
        mx[r] = -3.4e38f; se[r] = 0.0f; o0[r] = 0.0f; o1[r] = 0.0f;
    }
    // pass 1: row maxima
    for (int m = 0; m < Nn; ++m) {
        float2 kk = Ks[m];
#pragma unroll
        for (int r = 0; r < 8; ++r)
            mx[r] = fmaxf(mx[r], fmaf(q0[r], kk.x, q1[r] * kk.y) * scale);
    }
    // pass 2: exp-sum + weighted V
    for (int m = 0; m < Nn; ++m) {
        float2 kk = Ks[m];
        float2 vv = Vs[m];
#pragma unroll
        for (int r = 0; r < 8; ++r) {
            float e = __expf(fmaf(q0[r], kk.x, q1[r] * kk.y) * scale - mx[r]);
            se[r] += e;
            o0[r] = fmaf(e, vv.x, o0[r]);
            o1[r] = fmaf(e, vv.y, o1[r]);
        }
    }
    const float wo00 = Wo[s * 4], wo01 = Wo[s * 4 + 1];
    const float wo10 = Wo[s * 4 + 2], wo11 = Wo[s * 4 + 3];
    const float bo0 = bo[s * 2], bo1 = bo[s * 2 + 1];
    float asum = 0.0f;
#pragma unroll
    for (int r = 0; r < 8; ++r) {
        int n = r * 256 + tid;
        float O0 = q0[r] + o0[r] / se[r];
        float O1 = q1[r] + o1[r] / se[r];
        float t0 = fmaxf(fmaf(O0, wo00, fmaf(O1, wo10, bo0)), 0.0f);
        float t1 = fmaxf(fmaf(O0, wo01, fmaf(O1, wo11, bo1)), 0.0f);
        O0 += t0; O1 += t1;                      // mu = O0, alpha = O1
        size_t xi = ((size_t)(b * Nn + n)) * En + (s + 1);
        Zout[xi] = (X[xi] - O0) * __expf(-O1);
        asum += O1;
    }
    red[tid] = asum;
    __syncthreads();
    for (int off = 128; off > 0; off >>= 1) {
        if (tid < off) red[tid] += red[tid + off];
        __syncthreads();
    }
    if (tid == 0) alpha_ws[sb] = red[0];
}

// =====================================================================
// Kernel 3: Z[:,:,0] = (X[:,:,0] - mu0) * exp(-a0)
// =====================================================================
__global__ __launch_bounds__(256) void k3_z0(
    const float* __restrict__ X, const float* __restrict__ ip,
    float* __restrict__ Zout)
{
    int i = blockIdx.x * 256 + threadIdx.x;     // i = b*Nn + n
    float mu0 = ip[0], a0 = ip[1];
    Zout[(size_t)i * En] = (X[(size_t)i * En] - mu0) * __expf(-a0);
}

// =====================================================================
// Kernel 4: slogdet(convW)*N (16x16 LU, single lane) + logdet assembly.
// =====================================================================
__global__ __launch_bounds__(64) void k4_logdet(
    const float* __restrict__ convW, const float* __restrict__ ip,
    const float* __restrict__ alpha_ws, float* __restrict__ out_logdet)
{
    __shared__ float cl;
    if (threadIdx.x == 0) {
        float M[16][16];
        for (int r = 0; r < 16; ++r)
            for (int c = 0; c < 16; ++c) M[r][c] = convW[r * 16 + c];
        float lad = 0.0f;
        for (int k = 0; k < 16; ++k) {
            int p = k; float best = fabsf(M[k][k]);
            for (int i = k + 1; i < 16; ++i) {
                float v = fabsf(M[i][k]);
                if (v > best) { best = v; p = i; }
            }
            if (p != k)
                for (int j = 0; j < 16; ++j) {
                    float t = M[k][j]; M[k][j] = M[p][j]; M[p][j] = t;
                }
            float piv = M[k][k];
            lad += logf(fabsf(piv));
            for (int i = k + 1; i < 16; ++i) {
                float f = M[i][k] / piv;
                for (int j = k + 1; j < 16; ++j)
                    M[i][j] = fmaf(-f, M[k][j], M[i][j]);
            }
        }
        cl = lad * (float)Nn;
    }
    __syncthreads();
    int b = threadIdx.x;
    if (b < Bn) {
        float a0 = ip[1];
        float acc = -a0 * (float)Nn + cl;        // logdet0 + conv_logdet
        for (int s = 0; s < Sn; ++s) acc -= alpha_ws[s * Bn + b];
        out_logdet[b] = acc;
    }
}

extern "C" void kernel_launch(void* const* d_in, const int* in_sizes, int n_in,
                              void* d_out, int out_size, void* d_ws, size_t ws_size,
                              hipStream_t stream)
{
    const float* X   = (const float*)d_in[0];
    const float* W1  = (const float*)d_in[1];
    const float* b1  = (const float*)d_in[2];
    const float* W2  = (const float*)d_in[3];
    const float* b2  = (const float*)d_in[4];
    const float* W3  = (const float*)d_in[5];
    const float* b3  = (const float*)d_in[6];
    const float* Wq  = (const float*)d_in[7];
    const float* bq  = (const float*)d_in[8];
    const float* Wk  = (const float*)d_in[9];
    const float* bk  = (const float*)d_in[10];
    const float* Wv  = (const float*)d_in[11];
    const float* bvp = (const float*)d_in[12];
    const float* Wo  = (const float*)d_in[13];
    const float* bo  = (const float*)d_in[14];
    const float* cW  = (const float*)d_in[15];
    const float* ip  = (const float*)d_in[16];

    float* Z      = (float*)d_out;
    float* logdet = Z + (size_t)Bn * Nn * En;

    float* ws = (float*)d_ws;
    const size_t Nq = (size_t)Sn * Bn * Nn * 2;
    float* Qws = ws;
    float* Kws = ws + Nq;
    float* Vws = ws + 2 * Nq;
    float* aws = ws + 3 * Nq;

    dim3 g1(Nn / 64, Bn, Sn);
    k1_mlp_qkv<<<g1, 128, 0, stream>>>(X, W1, b1, W2, b2, W3, b3,
                                       Wq, bq, Wk, bk, Wv, bvp,
                                       Qws, Kws, Vws);
    k2_attn<<<Sn * Bn, 256, 0, stream>>>(X, Qws, Kws, Vws, Wo, bo, Z, aws);
    k3_z0<<<(Bn * Nn) / 256, 256, 0, stream>>>(X, ip, Z);
    k4_logdet<<<1, 64, 0, stream>>>(cW, ip, aws, logdet);
}